// Sub_MGAI2_7902739825288
// MI455X (gfx1250) — compile-verified
//
#include <hip/hip_runtime.h>
#include <hip/hip_bf16.h>
#include <math.h>

typedef __attribute__((ext_vector_type(2))) float v2f;
typedef __attribute__((ext_vector_type(8))) float v8f;

#define BB 8
#define CC 256
#define KK 8
#define HH 96
#define NN 9216                 // 96*96
#define CN (CC * NN)            // 2359296
#define KN (KK * NN)            // 73728
#define NCHUNKS 36              // 9216 / 256

// ---------------- workspace layout (float offsets) ----------------
#define OFF_SIGMA   0            // 2048
#define OFF_IS2     2048         // 2048
#define OFF_AIS2    4096         // 2048
#define OFF_CONSTK  6144         // 64 (8 used)
#define OFF_SCALE0  6208         // 256
#define OFF_SHIFT0  6464         // 256
#define OFF_SCALE1  6720         // 256
#define OFF_SHIFT1  6976         // 256
#define OFF_S       7232         // 64  (B*K)     } contiguous zero region
#define OFF_WX      7296         // 16384 (B*K*C) }
#define OFF_NODES   23680        // 16384
#define OFF_COEEX   40064        // 16384
#define OFF_WC      56448        // 16384
#define OFF_ASSIGN  72832        // 589824 (B*K*N)
// total = 662656 floats = ~2.65 MB

// ------------------------------------------------------------------
// Kernel 0: per-cluster params (sigma, 1/sigma^2, anchor/sigma^2,
// const_k = sum_c a^2/s^2) + folded BN scale/shift for both branches.
__global__ void k_params(const float* __restrict__ anchor, const float* __restrict__ sigma_p,
                         const float* __restrict__ g0, const float* __restrict__ be0,
                         const float* __restrict__ mn0, const float* __restrict__ vr0,
                         const float* __restrict__ g1, const float* __restrict__ be1,
                         const float* __restrict__ mn1, const float* __restrict__ vr1,
                         float* __restrict__ ws) {
    __shared__ float red[KK * CC];
    int t = threadIdx.x;
    float* sigma = ws + OFF_SIGMA;
    float* is2o  = ws + OFF_IS2;
    float* ais2  = ws + OFF_AIS2;
    float* ck    = ws + OFF_CONSTK;
#pragma unroll
    for (int k = 0; k < KK; ++k) {
        int i = k * CC + t;
        float sp = sigma_p[i];
        float s  = 1.0f / (1.0f + expf(-sp));
        sigma[i] = s;
        float is2 = 1.0f / (s * s);
        is2o[i] = is2;
        float a = anchor[i];
        ais2[i] = a * is2;
        red[i]  = a * a * is2;
    }
    __syncthreads();
    if (t < KK) {
        float s = 0.f;
        for (int i = 0; i < CC; ++i) s += red[t * CC + i];
        ck[t] = s;
    }
    float sc0 = g0[t] * rsqrtf(vr0[t] + 1e-5f);
    ws[OFF_SCALE0 + t] = sc0;
    ws[OFF_SHIFT0 + t] = be0[t] - mn0[t] * sc0;
    float sc1 = g1[t] * rsqrtf(vr1[t] + 1e-5f);
    ws[OFF_SCALE1 + t] = sc1;
    ws[OFF_SHIFT1 + t] = be1[t] - mn1[t] * sc1;
}

// Zero the S + wx accumulation region (16448 floats).
__global__ void k_zero(float* __restrict__ p, int n) {
    int i = blockIdx.x * 256 + threadIdx.x;
    if (i < n) p[i] = 0.f;
}

// ------------------------------------------------------------------
// Kernel 1: soft-assignment. One thread per pixel (coalesced over n):
// d2[k] = sum_c x^2*is2[k,c] - 2 x * (a*is2)[k,c] + const_k ; softmax over k.
// Also accumulates S[b,k] = sum_n assign via wave reduce + atomics.
__global__ void k_assign(const float* __restrict__ x, const float* __restrict__ ws,
                         float* __restrict__ assign, float* __restrict__ S) {
    __shared__ float ls_is2[KK * CC];
    __shared__ float ls_a2[KK * CC];
    int tid = threadIdx.x;
    for (int i = tid; i < KK * CC; i += 256) {
        ls_is2[i] = ws[OFF_IS2 + i];
        ls_a2[i]  = ws[OFF_AIS2 + i];
    }
    __syncthreads();
    int b = blockIdx.x / NCHUNKS;
    int n = (blockIdx.x % NCHUNKS) * 256 + tid;
    const float* xb = x + (size_t)b * CN + n;
    float acc[KK];
#pragma unroll
    for (int k = 0; k < KK; ++k) acc[k] = 0.f;
    for (int c = 0; c < CC; ++c) {
        float xv = xb[(size_t)c * NN];
        float p = xv * xv;
        float q = -2.0f * xv;
#pragma unroll
        for (int k = 0; k < KK; ++k)
            acc[k] = fmaf(q, ls_a2[k * CC + c], fmaf(p, ls_is2[k * CC + c], acc[k]));
    }
    float lg[KK];
    float mx = -3.4e38f;
#pragma unroll
    for (int k = 0; k < KK; ++k) {
        lg[k] = -0.5f * (acc[k] + ws[OFF_CONSTK + k]);
        mx = fmaxf(mx, lg[k]);
    }
    float sum = 0.f;
#pragma unroll
    for (int k = 0; k < KK; ++k) { lg[k] = __expf(lg[k] - mx); sum += lg[k]; }
    float inv = 1.0f / sum;
#pragma unroll
    for (int k = 0; k < KK; ++k) {
        float a = lg[k] * inv;
        lg[k] = a;
        assign[(size_t)b * KN + (size_t)k * NN + n] = a;
    }
    // wave32 reduce, then one atomic per wave per cluster
#pragma unroll
    for (int k = 0; k < KK; ++k) {
        float v = lg[k];
        for (int off = 16; off > 0; off >>= 1) v += __shfl_xor(v, off, 32);
        if ((tid & 31) == 0) atomicAdd(&S[b * KK + k], v);
    }
}

// ------------------------------------------------------------------
// Kernel 2: wx[b,k,c] = sum_n assign[b,k,n] * x[b,c,n] (partial per n-chunk).
__global__ void k_wx(const float* __restrict__ x, const float* __restrict__ assign,
                     float* __restrict__ wx) {
    __shared__ float ls_a[KK * 256];
    int tid = threadIdx.x;
    int b = blockIdx.x / NCHUNKS;
    int n0 = (blockIdx.x % NCHUNKS) * 256;
#pragma unroll
    for (int k = 0; k < KK; ++k)
        ls_a[k * 256 + tid] = assign[(size_t)b * KN + (size_t)k * NN + n0 + tid];
    __syncthreads();
    const float* xr = x + (size_t)b * CN + (size_t)tid * NN + n0;
    float acc[KK];
#pragma unroll
    for (int k = 0; k < KK; ++k) acc[k] = 0.f;
    for (int i = 0; i < 256; ++i) {
        float xv = xr[i];
#pragma unroll
        for (int k = 0; k < KK; ++k) acc[k] = fmaf(ls_a[k * 256 + i], xv, acc[k]);
    }
#pragma unroll
    for (int k = 0; k < KK; ++k)
        atomicAdd(&wx[b * (KK * CC) + k * CC + tid], acc[k]);
}

// ------------------------------------------------------------------
// Kernel 3: nodes = (wx - S*anchor)/sigma/(S+eps); per-node L2 norm; global L2.
// Output kept flat in [K,C] order (graph view [C,K] = flat[c*8+k]).
__global__ void k_nodes(const float* __restrict__ ws_wx, const float* __restrict__ Sv,
                        const float* __restrict__ sigma, const float* __restrict__ anchor,
                        float* __restrict__ nodes) {
    __shared__ float vbuf[KK * CC];
    __shared__ float ssum[KK], sinv[KK];
    int b = blockIdx.x, t = threadIdx.x;
    float v[KK];
#pragma unroll
    for (int k = 0; k < KK; ++k) {
        float Sk = Sv[b * KK + k];
        int i = k * CC + t;
        float val = (ws_wx[b * (KK * CC) + i] - Sk * anchor[i]) / sigma[i] / (Sk + 1e-9f);
        v[k] = val;
        vbuf[i] = val * val;
    }
    __syncthreads();
    if (t < KK) {
        float s = 0.f;
        for (int i = 0; i < CC; ++i) s += vbuf[t * CC + i];
        ssum[t] = s;
        sinv[t] = 1.0f / fmaxf(sqrtf(s), 1e-12f);
    }
    __syncthreads();
    float g2 = 0.f;
#pragma unroll
    for (int k = 0; k < KK; ++k) g2 += ssum[k] * sinv[k] * sinv[k];
    float ginv = 1.0f / fmaxf(sqrtf(g2), 1e-12f);
#pragma unroll
    for (int k = 0; k < KK; ++k)
        nodes[b * (KK * CC) + k * CC + t] = v[k] * sinv[k] * ginv;
}

// ------------------------------------------------------------------
// Kernel 4: CascadeGCNet (3 iterations), K=8 so everything lives in LDS.
__global__ void k_cascade(const float* __restrict__ nodes,
                          const float* __restrict__ gw1, const float* __restrict__ gw2,
                          const float* __restrict__ gw3, float* __restrict__ coeex) {
    __shared__ float xl[CC * KK];   // [c][k]
    __shared__ float gram[64];
    __shared__ float adj[64];
    int b = blockIdx.x, t = threadIdx.x;
    for (int i = t; i < CC * KK; i += 256) xl[i] = nodes[b * (KK * CC) + i]; // raw .view copy
    __syncthreads();
    const float* Ws[3] = {gw1, gw2, gw3};
    for (int it = 0; it < 3; ++it) {
        if (t < 64) {   // gram[i][j] = sum_c x[c,i]*x[c,j]
            int i = t >> 3, j = t & 7;
            float s = 0.f;
            for (int c = 0; c < CC; ++c) s += xl[c * KK + i] * xl[c * KK + j];
            gram[t] = s;
        }
        __syncthreads();
        if (t < KK) {   // softmax over rows
            float mx = -3.4e38f;
            for (int j = 0; j < KK; ++j) mx = fmaxf(mx, gram[t * KK + j]);
            float e[KK]; float sm = 0.f;
            for (int j = 0; j < KK; ++j) { e[j] = __expf(gram[t * KK + j] - mx); sm += e[j]; }
            float inv = 1.0f / sm;
            for (int j = 0; j < KK; ++j) adj[t * KK + j] = e[j] * inv;
        }
        __syncthreads();
        // y[k] for c_out = t: sum_ci x[ci,k] * Wm[ci, t]
        float y[KK];
#pragma unroll
        for (int k = 0; k < KK; ++k) y[k] = 0.f;
        const float* Wm = Ws[it];
        for (int ci = 0; ci < CC; ++ci) {
            float w = Wm[ci * CC + t];
#pragma unroll
            for (int k = 0; k < KK; ++k) y[k] = fmaf(xl[ci * KK + k], w, y[k]);
        }
        float xn[KK];
#pragma unroll
        for (int k = 0; k < KK; ++k) {
            float s = 0.f;
#pragma unroll
            for (int j = 0; j < KK; ++j) s = fmaf(adj[k * KK + j], y[j], s);
            xn[k] = s;
        }
        __syncthreads();
#pragma unroll
        for (int k = 0; k < KK; ++k) xl[t * KK + k] = xn[k];
        __syncthreads();
    }
#pragma unroll
    for (int k = 0; k < KK; ++k)
        coeex[b * (KK * CC) + t * KK + k] = fmaxf(xl[t * KK + k], 0.f);
}

// ------------------------------------------------------------------
// Kernel 5: wc[b,o,k] = sum_c W[o,c] * coeex[b,c,k]  via V_WMMA_F32_16X16X4_F32.
// coeex is staged in LDS transposed to [k][c] so each B fetch is one ds_load_b64.
// Lanes 8..15 compute duplicate columns (kcol = m&7): no divergent loads, no
// EXEC churn around the WMMAs; only columns < 8 are stored.
__global__ void k_wc(const float* __restrict__ W, const float* __restrict__ coeex,
                     float* __restrict__ wc) {
    __shared__ float lsT[KK * CC];    // [k][c]
    int tid = threadIdx.x;
    int b = blockIdx.x >> 1, half = blockIdx.x & 1;
    const float* cb = coeex + b * (KK * CC);
    for (int i = tid; i < KK * CC; i += 256)
        lsT[(i & 7) * CC + (i >> 3)] = cb[i];   // coalesced global read, [c][k] -> [k][c]
    __syncthreads();

    int wave = tid >> 5, lane = tid & 31;
    int o0 = (half * 8 + wave) * 16;
    int m = lane & 15, kh = lane >> 4;
    int kcol = m & 7;                            // duplicate cluster column for lanes 8..15
    const float* Wrow = W + (size_t)(o0 + m) * CC + 2 * kh;
    const float* Brow = lsT + kcol * CC + 2 * kh;

    v8f acc0 = {}, acc1 = {};
    for (int c0 = 0; c0 < CC; c0 += 8) {
        v2f a0 = *(const v2f*)(Wrow + c0);
        v2f b0 = *(const v2f*)(Brow + c0);
        acc0 = __builtin_amdgcn_wmma_f32_16x16x4_f32(false, a0, false, b0, (short)0, acc0,
                                                     false, false);
        v2f a1 = *(const v2f*)(Wrow + c0 + 4);
        v2f b1 = *(const v2f*)(Brow + c0 + 4);
        acc1 = __builtin_amdgcn_wmma_f32_16x16x4_f32(false, a1, false, b1, (short)0, acc1,
                                                     false, false);
    }
    if (m < KK) {
#pragma unroll
        for (int r = 0; r < 8; ++r) {
            int o = o0 + r + 8 * kh;
            wc[b * (KK * CC) + o * KK + m] = acc0[r] + acc1[r];
        }
    }
}

// ------------------------------------------------------------------
// Kernel 6: out[b,o,n] = x[b,o,n] + relu( (wc@assign)*scale + shift )
// 16x16 output tile per wave, two K=4 WMMA steps cover the 8 clusters.
__global__ void k_final(const float* __restrict__ wc, const float* __restrict__ assign,
                        const float* __restrict__ x, const float* __restrict__ scale,
                        const float* __restrict__ shift, float* __restrict__ out) {
    int tid = threadIdx.x;
    int wave = tid >> 5, lane = tid & 31;
    long id = (long)blockIdx.x * 8 + wave;
    int ntile = (int)(id % 576);
    long r1 = id / 576;
    int otile = (int)(r1 % 16);
    int b = (int)(r1 / 16);
    int n0 = ntile * 16, o0 = otile * 16;
    int m = lane & 15, kh = lane >> 4;

    const float* wcb = wc + b * (KK * CC) + (o0 + m) * KK + 2 * kh;
    v2f a0 = *(const v2f*)(wcb);          // K = 2kh, 2kh+1
    v2f a1 = *(const v2f*)(wcb + 4);      // K = 4+2kh, 5+2kh
    const float* ab = assign + (size_t)b * KN + n0 + m;
    v2f b0, b1;
    b0.x = ab[(size_t)(2 * kh) * NN];     b0.y = ab[(size_t)(2 * kh + 1) * NN];
    b1.x = ab[(size_t)(4 + 2 * kh) * NN]; b1.y = ab[(size_t)(5 + 2 * kh) * NN];

    v8f acc = {};
    acc = __builtin_amdgcn_wmma_f32_16x16x4_f32(false, a0, false, b0, (short)0, acc, false, false);
    acc = __builtin_amdgcn_wmma_f32_16x16x4_f32(false, a1, false, b1, (short)0, acc, false, false);

    int ncol = n0 + m;
    const float* xb = x + (size_t)b * CN + ncol;
    float* ob = out + (size_t)b * CN + ncol;
#pragma unroll
    for (int r = 0; r < 8; ++r) {
        int o = o0 + r + 8 * kh;
        float v = fmaf(acc[r], scale[o], shift[o]);
        v = fmaxf(v, 0.f);
        ob[(size_t)o * NN] = xb[(size_t)o * NN] + v;
    }
}

// ------------------------------------------------------------------
extern "C" void kernel_launch(void* const* d_in, const int* in_sizes, int n_in,
                              void* d_out, int out_size, void* d_ws, size_t ws_size,
                              hipStream_t stream) {
    const float* edger  = (const float*)d_in[0];
    const float* edget  = (const float*)d_in[1];
    const float* anchor = (const float*)d_in[2];
    const float* sigmap = (const float*)d_in[3];
    const float* gw1 = (const float*)d_in[4];
    const float* gw2 = (const float*)d_in[5];
    const float* gw3 = (const float*)d_in[6];
    const float* c0_w = (const float*)d_in[7];
    const float* c0_g = (const float*)d_in[8];
    const float* c0_b = (const float*)d_in[9];
    const float* c0_m = (const float*)d_in[10];
    const float* c0_v = (const float*)d_in[11];
    const float* c1_w = (const float*)d_in[12];
    const float* c1_g = (const float*)d_in[13];
    const float* c1_b = (const float*)d_in[14];
    const float* c1_m = (const float*)d_in[15];
    const float* c1_v = (const float*)d_in[16];

    float* ws = (float*)d_ws;
    float* Sbuf   = ws + OFF_S;
    float* wx     = ws + OFF_WX;
    float* nodes  = ws + OFF_NODES;
    float* coeex  = ws + OFF_COEEX;
    float* wc     = ws + OFF_WC;
    float* assign = ws + OFF_ASSIGN;

    k_params<<<1, 256, 0, stream>>>(anchor, sigmap, c0_g, c0_b, c0_m, c0_v,
                                    c1_g, c1_b, c1_m, c1_v, ws);

    for (int br = 0; br < 2; ++br) {
        const float* x  = br ? edget : edger;
        const float* cw = br ? c1_w : c0_w;
        const float* scale = ws + (br ? OFF_SCALE1 : OFF_SCALE0);
        const float* shift = ws + (br ? OFF_SHIFT1 : OFF_SHIFT0);
        float* outp = (float*)d_out + (size_t)br * BB * CN;

        k_zero<<<65, 256, 0, stream>>>(ws + OFF_S, 64 + BB * KK * CC);
        k_assign<<<BB * NCHUNKS, 256, 0, stream>>>(x, ws, assign, Sbuf);
        k_wx<<<BB * NCHUNKS, 256, 0, stream>>>(x, assign, wx);
        k_nodes<<<BB, 256, 0, stream>>>(wx, Sbuf, ws + OFF_SIGMA, anchor, nodes);
        k_cascade<<<BB, 256, 0, stream>>>(nodes, gw1, gw2, gw3, coeex);
        k_wc<<<2 * BB, 256, 0, stream>>>(cw, coeex, wc);
        k_final<<<BB * 16 * 576 / 8, 256, 0, stream>>>(wc, assign, x, scale, shift, outp);
    }
}